// Lora_48936857371143
// MI455X (gfx1250) — compile-verified
//
#include <hip/hip_runtime.h>

typedef __attribute__((ext_vector_type(2))) float v2f;
typedef __attribute__((ext_vector_type(8))) float v8f;

#define K_DIM 4096
#define N_DIM 4096
#define R_DIM 8
#define ROWS_PER_BLOCK 16
#define WAVES_PER_BLOCK 8
#define THREADS_PER_BLOCK 256
#define LORA_SCALE 2.0f

// Stage-1 x staging in LDS: per wave, double-buffered 16 rows x 32 k-values.
// Row stride padded to 34 words: keeps 8B alignment for ds_load_b64 A-operand
// reads (34*r even + even k offsets) and spreads rows across banks.
#define CHUNK_K    32
#define ROW_WORDS  34
#define BUF_WORDS  (ROWS_PER_BLOCK * ROW_WORDS)       // 544
#define WAVE_WORDS (2 * BUF_WORDS)                    // 1088 (double buffer)
#define NCHUNK     ((K_DIM / WAVES_PER_BLOCK) / CHUNK_K)  // 16
#define ASYNC_PER_CHUNK 4                             // 4 x 512B = 16x32 floats

__device__ __forceinline__ void async_copy_b128(unsigned lds_byte, const void* gaddr) {
    // per-lane 16B global -> LDS, tracked with ASYNCcnt
    asm volatile("global_load_async_to_lds_b128 %0, %1, off"
                 :: "v"(lds_byte), "v"(gaddr) : "memory");
}

// Issue one 16-row x 32-float chunk: 4 async instrs, each 4 rows x 128B, coalesced.
__device__ __forceinline__ void issue_chunk(const float* gbase, unsigned lds_byte) {
    #pragma unroll
    for (int j = 0; j < ASYNC_PER_CHUNK; ++j)
        async_copy_b128(lds_byte + (unsigned)(j * 4 * ROW_WORDS * 4),
                        gbase + (size_t)(4 * j) * K_DIM);
}

__global__ __launch_bounds__(THREADS_PER_BLOCK)
void lora_fused_wmma_kernel(const float* __restrict__ x,
                            const float* __restrict__ LA,   // [4096, 8]
                            const float* __restrict__ LB,   // [8, 4096]
                            float* __restrict__ y)
{
    __shared__ float lds_x[WAVES_PER_BLOCK * WAVE_WORDS];   // ~34 KB
    __shared__ float lds_part[WAVES_PER_BLOCK * 256];       // 8 KB partial C tiles
    __shared__ float lds_t[ROWS_PER_BLOCK * R_DIM];         // T = (x_tile @ LA) * scale

    const int tid  = threadIdx.x;
    const int wave = tid >> 5;
    const int lane = tid & 31;
    const int row_base = blockIdx.x * ROWS_PER_BLOCK;

    const int rowL = lane & 15;        // A-op row / B-op column
    const int kHi  = (lane >> 4) * 2;  // K sub-offset: lanes 0-15 -> {k,k+1}, 16-31 -> {k+2,k+3}

    const int k_begin = wave * (K_DIM / WAVES_PER_BLOCK);

    // ---- Stage 1: T = x_tile @ LA via WMMA, x staged through LDS with async copies ----
    // Staging addresses: async instr j covers rows 4j..4j+3; lane l -> row 4j+(l>>3),
    // floats (l&7)*4 .. +3 of the 32-float chunk.
    const unsigned lds_x_base = (unsigned)(size_t)&lds_x[0];
    const unsigned lane_word  = (unsigned)(wave * WAVE_WORDS
                              + (lane >> 3) * ROW_WORDS + (lane & 7) * 4);
    const unsigned lbyte0 = lds_x_base + 4u * lane_word;
    const unsigned lbyte1 = lbyte0 + 4u * BUF_WORDS;
    const float* gchunk = x + (size_t)(row_base + (lane >> 3)) * K_DIM
                            + (lane & 7) * 4 + k_begin;

    issue_chunk(gchunk, lbyte0);                 // chunk 0 -> buf 0
    issue_chunk(gchunk + CHUNK_K, lbyte1);       // chunk 1 -> buf 1

    // B-operand columns 8..15 of the C tile are discarded by the reduction, so
    // lanes 8..15 just load duplicates of columns 0..7 — no mask, no EXEC branch.
    const int bcol = rowL & (R_DIM - 1);
    // Single per-wave base pointer; inner-loop LA accesses become immediate offsets.
    const float* lap = LA + (size_t)(k_begin + kHi) * R_DIM + bcol;
    const unsigned rd_word = (unsigned)(wave * WAVE_WORDS + rowL * ROW_WORDS + kHi);

    v8f acc = {};
    for (int i = 0; i < NCHUNK; ++i) {
        // In-order async completion: <=4 outstanding => chunk i landed.
        if (i + 1 < NCHUNK)
            asm volatile("s_wait_asynccnt 0x4" ::: "memory");
        else
            asm volatile("s_wait_asynccnt 0x0" ::: "memory");

        const unsigned base = rd_word + (unsigned)(i & 1) * BUF_WORDS;
        #pragma unroll
        for (int kc = 0; kc < CHUNK_K; kc += 4) {
            v2f a = *(const v2f*)&lds_x[base + kc];     // ds_load_b64, 8B aligned
            v2f b;
            b.x = lap[(size_t)(kc)     * R_DIM];        // global_load_b32 offset:kc*32
            b.y = lap[(size_t)(kc + 1) * R_DIM];        // global_load_b32 offset:kc*32+32
            acc = __builtin_amdgcn_wmma_f32_16x16x4_f32(false, a, false, b,
                                                        (short)0, acc, false, false);
        }
        lap += (size_t)CHUNK_K * R_DIM;
        if (i + 2 < NCHUNK)   // reuse the buffer we just finished reading
            issue_chunk(gchunk + (size_t)(i + 2) * CHUNK_K,
                        (i & 1) ? lbyte1 : lbyte0);
    }

    // Per-wave partials -> LDS
    #pragma unroll
    for (int i = 0; i < 8; ++i)
        lds_part[wave * 256 + lane * 8 + i] = acc[i];
    __syncthreads();

    // Cross-wave reduction; fold LoRA scale into T (cols 8..15 of C are dropped here)
    {
        float s = 0.0f;
        #pragma unroll
        for (int w = 0; w < WAVES_PER_BLOCK; ++w)
            s += lds_part[w * 256 + tid];
        const int l = tid >> 3;                 // owning lane in C layout
        const int i = tid & 7;                  // C VGPR index
        const int m = (l < 16) ? i : (i + 8);   // C row
        const int n = (l < 16) ? l : (l - 16);  // C col (only 0..7 meaningful)
        if (n < R_DIM)
            lds_t[m * R_DIM + n] = s * LORA_SCALE;
    }
    __syncthreads();

    // ---------------- Stage 2: y_tile = T @ LB via two chained WMMAs ----------------
    v2f ta0, ta1;
    ta0.x = lds_t[rowL * R_DIM + kHi];
    ta0.y = lds_t[rowL * R_DIM + kHi + 1];
    ta1.x = lds_t[rowL * R_DIM + 4 + kHi];
    ta1.y = lds_t[rowL * R_DIM + 4 + kHi + 1];

    const float* lbp = LB + (size_t)kHi * N_DIM;   // rows kHi / kHi+1 / kHi+4 / kHi+5 via imm offsets
    for (int n0 = wave * 16; n0 < N_DIM; n0 += WAVES_PER_BLOCK * 16) {
        const int col = n0 + rowL;
        v2f b0, b1;
        b0.x = lbp[col];
        b0.y = lbp[col + (size_t)N_DIM];
        b1.x = lbp[col + (size_t)4 * N_DIM];
        b1.y = lbp[col + (size_t)5 * N_DIM];

        v8f c = {};
        c = __builtin_amdgcn_wmma_f32_16x16x4_f32(false, ta0, false, b0,
                                                  (short)0, c, false, false);
        c = __builtin_amdgcn_wmma_f32_16x16x4_f32(false, ta1, false, b1,
                                                  (short)0, c, false, false);

        // D layout: VGPR i -> row i (lanes 0-15) / row i+8 (lanes 16-31)
        const int rbase = row_base + ((lane < 16) ? 0 : 8);
        float* ybase = y + (size_t)rbase * N_DIM + col;
        #pragma unroll
        for (int i = 0; i < 8; ++i)
            ybase[(size_t)i * N_DIM] = c[i];
    }
}

extern "C" void kernel_launch(void* const* d_in, const int* in_sizes, int n_in,
                              void* d_out, int out_size, void* d_ws, size_t ws_size,
                              hipStream_t stream) {
    (void)n_in; (void)out_size; (void)d_ws; (void)ws_size;
    const float* x  = (const float*)d_in[0];   // [4, 2048, 4096]
    const float* LA = (const float*)d_in[1];   // [4096, 8]
    const float* LB = (const float*)d_in[2];   // [8, 4096]
    float* y = (float*)d_out;                  // [4, 2048, 4096]

    const int M = in_sizes[0] / K_DIM;         // 8192 rows
    const int blocks = M / ROWS_PER_BLOCK;     // 512
    hipLaunchKernelGGL(lora_fused_wmma_kernel, dim3(blocks), dim3(THREADS_PER_BLOCK),
                       0, stream, x, LA, LB, y);
}